// DigitCapsLayer_1872605741921
// MI455X (gfx1250) — compile-verified
//
#include <hip/hip_runtime.h>
#include <hip/hip_bf16.h>
#include <math.h>

#define B_ 32
#define M_ 32
#define P_ 196
#define D_ 20
#define O_ 16
#define I_ 8
#define DO_ (D_*O_)   // 320
#define PD_ (P_*D_)   // 3920

typedef __attribute__((ext_vector_type(2))) float v2f;
typedef __attribute__((ext_vector_type(4))) float v4f;
typedef __attribute__((ext_vector_type(8))) float v8f;
typedef __attribute__((ext_vector_type(4))) unsigned int u32x4;
typedef __attribute__((ext_vector_type(4))) int i32x4;
typedef __attribute__((ext_vector_type(8))) int i32x8;

// ---------------------------------------------------------------------------
// u_hat[b,m,p,do] = sum_i W[m,p,do,i] * x[b,m,p,i]
// One block per (m,p). 128 threads = 4 waves. 20 (do) x 2 (b) tiles of 16x16,
// K = I = 8 as two chained V_WMMA_F32_16X16X4_F32.
// ---------------------------------------------------------------------------
__global__ void uhat_gemm_kernel(const float* __restrict__ x,
                                 const float* __restrict__ W,
                                 float* __restrict__ uh) {
  const int mp   = blockIdx.x;          // 0..M*P-1
  const int m    = mp / P_;
  const int lane = threadIdx.x & 31;
  const int wave = threadIdx.x >> 5;
  const int half = lane >> 4;           // 0: K=0,1  1: K=2,3
  const int l16  = lane & 15;
  const int k0   = half * 2;

  const float* Wp = W + (size_t)mp * DO_ * I_;

  for (int t = wave; t < 40; t += 4) {
    const int mt = t >> 1;              // do tile 0..19
    const int nt = t & 1;               // b  tile 0..1
    const int r  = mt * 16 + l16;       // do row for A fragment
    const float* Wr = Wp + (size_t)r * I_;

    // Branchless speculative prefetch of the next tile's A rows
    // (global_prefetch_b8; clamp keeps the address in-bounds, no exec mask ops)
    const int tn = (t + 4 < 40) ? (t + 4) : t;
    const int r2 = (tn >> 1) * 16 + l16;
    __builtin_prefetch(Wp + (size_t)r2 * I_, 0, 1);

    v2f a0 = *(const v2f*)(Wr + k0);        // K = k0, k0+1
    v2f a1 = *(const v2f*)(Wr + k0 + 4);    // K = k0+4, k0+5
    const int bcol = nt * 16 + l16;     // batch column for B fragment
    const float* xr = x + (((size_t)bcol * M_ + m) * P_ + (mp % P_)) * I_;
    v2f b0 = *(const v2f*)(xr + k0);
    v2f b1 = *(const v2f*)(xr + k0 + 4);

    v8f c = {};
    c = __builtin_amdgcn_wmma_f32_16x16x4_f32(false, a0, false, b0, (short)0, c, false, false);
    c = __builtin_amdgcn_wmma_f32_16x16x4_f32(false, a1, false, b1, (short)0, c, false, false);

    // D vgpr j = row (half*8 + j), col = bcol -> contiguous 8 floats per lane
    const size_t base = (((size_t)bcol * M_ + m) * P_ + (mp % P_)) * DO_
                        + (size_t)mt * 16 + half * 8;
    *(v8f*)(uh + base) = c;
  }
}

// ---------------------------------------------------------------------------
// s0[b,do] = (1/(P*D)) * sum_{m,p} u_hat[b,m,p,do]   (iteration 0, uniform c)
// ---------------------------------------------------------------------------
__global__ void s0_reduce_kernel(const float* __restrict__ uh,
                                 float* __restrict__ s0) {
  const int bm  = blockIdx.x;
  const int b   = bm / M_;
  const int dox = threadIdx.x;          // 0..319
  const float* base = uh + (size_t)bm * P_ * DO_;
  float acc = 0.f;
  for (int p = 0; p < P_; ++p) acc += base[(size_t)p * DO_ + dox];
  atomicAdd(&s0[b * DO_ + dox], acc * (1.0f / (float)PD_));
}

// ---------------------------------------------------------------------------
// Stage one (b,m) u_hat chunk (250,880 B contiguous) into LDS via the Tensor
// Data Mover: wave 0 issues a 1-D tensor_load_to_lds, waits TENSORcnt==0,
// then the workgroup barrier publishes it to all waves.
// ---------------------------------------------------------------------------
__device__ __forceinline__ void stage_uh_tdm(const float* __restrict__ g,
                                             float* __restrict__ uh,
                                             int tid, int nthreads) {
#if __has_builtin(__builtin_amdgcn_tensor_load_to_lds) && __has_builtin(__builtin_amdgcn_s_wait_tensorcnt)
  if (tid < 32) {  // one wave issues the DMA (TDM ignores EXEC; wave-level op)
    const unsigned long long ga = (unsigned long long)(uintptr_t)g;
    const unsigned int lds_off = (unsigned int)(uintptr_t)uh;  // LDS aperture: low 32b = offset
    const unsigned int N = (unsigned int)(P_ * DO_);           // 62720 elems of 4B

    u32x4 g0;
    g0[0] = 1u;                                                // count=1, user desc
    g0[1] = lds_off;                                           // lds_addr
    g0[2] = (unsigned int)(ga & 0xFFFFFFFFu);                  // global_addr[31:0]
    g0[3] = (unsigned int)((ga >> 32) & 0x01FFFFFFu) | (2u << 30); // addr[56:32] | type=2

    i32x8 g1;
    g1[0] = (int)(2u << 16);                 // workgroup_mask=0, data_size=2 (4B)
    g1[1] = (int)((N & 0xFFFFu) << 16);      // tensor_dim0[15:0] (bits 63:48)
    g1[2] = (int)((N >> 16) & 0xFFFFu) | (int)(1u << 16); // tensor_dim0 hi | tensor_dim1=1
    g1[3] = (int)((N & 0xFFFFu) << 16);      // tile_dim0 = N (bits 127:112)
    g1[4] = 1;                               // tile_dim1=1, tile_dim2=0
    g1[5] = (int)N;                          // tensor_dim0_stride[31:0]
    g1[6] = 0;                               // stride hi / dim1_stride lo
    g1[7] = 0;

    i32x4 zz = {0, 0, 0, 0};
#if __clang_major__ >= 23
    i32x8 z8 = {0, 0, 0, 0, 0, 0, 0, 0};
    __builtin_amdgcn_tensor_load_to_lds(g0, g1, zz, zz, z8, 0);
#else
    __builtin_amdgcn_tensor_load_to_lds(g0, g1, zz, zz, 0);
#endif
    __builtin_amdgcn_s_wait_tensorcnt(0);
  }
#else
  for (int i = tid; i < P_ * DO_ / 4; i += nthreads)
    ((v4f*)uh)[i] = ((const v4f*)g)[i];
#endif
}

// ---------------------------------------------------------------------------
// One routing pass for block (b,m), LDS-resident u_hat chunk (250.9 KB):
//   logits[pd] = sum_o uh[p,do]*vin[b,do] ; c = softmax_pd(logits)
//   s_acc[b,do] += sum_p c[p,d]*uh[p,do]
// final pass additionally writes c_maps and features.
// ---------------------------------------------------------------------------
__global__ void routing_kernel(const float* __restrict__ uh_g,
                               const float* __restrict__ vin,   // [B,DO]
                               float* __restrict__ s_acc,       // [B,DO]
                               float* __restrict__ cmaps,       // [B,M,P,D] (final)
                               float* __restrict__ feats,       // [B,M,D,O] (final)
                               int final_pass) {
  extern __shared__ float smem[];
  float* uh  = smem;                    // P*DO = 62720
  float* cc  = uh + P_ * DO_;           // PD   = 3920  (logits then c)
  float* vv  = cc + PD_;                // DO   = 320
  float* red = vv + DO_;                // 320 reduction scratch

  const int bm  = blockIdx.x;
  const int b   = bm / M_;
  const int tid = threadIdx.x;          // 0..319
  const float* g = uh_g + (size_t)bm * P_ * DO_;

  stage_uh_tdm(g, uh, tid, (int)blockDim.x);
  vv[tid] = vin[b * DO_ + tid];
  __syncthreads();

  // logits + running max
  float lmax = -3.0e38f;
  for (int pd = tid; pd < PD_; pd += blockDim.x) {
    const int p = pd / D_, d = pd % D_;
    const float* u = uh + p * DO_ + d * O_;
    const float* v = vv + d * O_;
    float acc = 0.f;
#pragma unroll
    for (int o = 0; o < O_; ++o) acc += u[o] * v[o];
    cc[pd] = acc;
    lmax = fmaxf(lmax, acc);
  }
  red[tid] = lmax;
  __syncthreads();
  if (tid == 0) {
    float mx = -3.0e38f;
    for (int i = 0; i < (int)blockDim.x; ++i) mx = fmaxf(mx, red[i]);
    red[0] = mx;
  }
  __syncthreads();
  const float mx = red[0];
  __syncthreads();

  float lsum = 0.f;
  for (int pd = tid; pd < PD_; pd += blockDim.x) {
    const float e = __expf(cc[pd] - mx);
    cc[pd] = e;
    lsum += e;
  }
  red[tid] = lsum;
  __syncthreads();
  if (tid == 0) {
    float s = 0.f;
    for (int i = 0; i < (int)blockDim.x; ++i) s += red[i];
    red[0] = s;
  }
  __syncthreads();
  const float inv = 1.0f / red[0];
  for (int pd = tid; pd < PD_; pd += blockDim.x) cc[pd] *= inv;
  __syncthreads();

  // partial s over p for this (b,m); doubles as features on final pass
  {
    const int dox = tid;
    const int d = dox >> 4;
    float acc = 0.f;
    for (int p = 0; p < P_; ++p) acc += cc[p * D_ + d] * uh[p * DO_ + dox];
    atomicAdd(&s_acc[b * DO_ + dox], acc);
    if (final_pass) feats[(size_t)bm * DO_ + dox] = acc * (1.0f / (float)P_);
  }

  if (final_pass) {
    for (int pd = tid; pd < PD_; pd += blockDim.x) {
      const int p = pd / D_, d = pd % D_;
      const float* u = uh + p * DO_ + d * O_;
      float sq = 0.f;
#pragma unroll
      for (int o = 0; o < O_; ++o) sq += u[o] * u[o];
      cmaps[(size_t)bm * PD_ + pd] = cc[pd] * sqrtf(sq);  // c >= 0 (softmax)
    }
  }
}

// ---------------------------------------------------------------------------
// vout[b,do] = (vprev ? vprev : 0) + squash(s)[b,do]
// ---------------------------------------------------------------------------
__global__ void squash_kernel(const float* __restrict__ s,
                              const float* __restrict__ vprev,
                              float* __restrict__ vout) {
  __shared__ float sh[DO_];
  const int b = blockIdx.x;
  const int t = threadIdx.x;            // 0..319
  const float val = s[b * DO_ + t];
  sh[t] = val * val;
  __syncthreads();
  const int d = t >> 4;
  float sq = 0.f;
#pragma unroll
  for (int o = 0; o < O_; ++o) sq += sh[d * O_ + o];
  float out = (sq / (1.0f + sq)) * val * rsqrtf(sq + 1e-9f);
  if (vprev) out += vprev[b * DO_ + t];
  vout[b * DO_ + t] = out;
}

__global__ void zero_kernel(float* __restrict__ p, int n) {
  const int i = blockIdx.x * blockDim.x + threadIdx.x;
  if (i < n) p[i] = 0.f;
}

// ---------------------------------------------------------------------------
extern "C" void kernel_launch(void* const* d_in, const int* in_sizes, int n_in,
                              void* d_out, int out_size, void* d_ws, size_t ws_size,
                              hipStream_t stream) {
  (void)in_sizes; (void)n_in; (void)out_size; (void)ws_size;
  const float* x = (const float*)d_in[0];   // [B,M,P,I]
  const float* W = (const float*)d_in[1];   // [1,M,P,D,O,I]
  float* out = (float*)d_out;

  float* ws = (float*)d_ws;
  const size_t UH = (size_t)B_ * M_ * P_ * DO_;   // 64,225,280 floats
  float* uh   = ws;
  float* s0   = uh + UH;
  float* s1   = s0 + B_ * DO_;
  float* s2   = s1 + B_ * DO_;
  float* v0   = s2 + B_ * DO_;
  float* vsum = v0 + B_ * DO_;

  float* out_v     = out;                                   // [B,D,O]
  float* out_cmaps = out + B_ * DO_;                        // [B,M,14,14,D]
  float* out_feats = out_cmaps + (size_t)B_ * M_ * PD_;     // [B,M,D,O]

  const size_t shmem = (size_t)(P_ * DO_ + PD_ + DO_ + DO_) * sizeof(float); // 269,120 B

  zero_kernel<<<(3 * B_ * DO_ + 255) / 256, 256, 0, stream>>>(s0, 3 * B_ * DO_);
  uhat_gemm_kernel<<<M_ * P_, 128, 0, stream>>>(x, W, uh);
  s0_reduce_kernel<<<B_ * M_, DO_, 0, stream>>>(uh, s0);
  squash_kernel<<<B_, DO_, 0, stream>>>(s0, nullptr, v0);                       // v0
  routing_kernel<<<B_ * M_, DO_, shmem, stream>>>(uh, v0, s1, nullptr, nullptr, 0);
  squash_kernel<<<B_, DO_, 0, stream>>>(s1, v0, vsum);                          // v0+v1
  routing_kernel<<<B_ * M_, DO_, shmem, stream>>>(uh, vsum, s2, out_cmaps, out_feats, 1);
  squash_kernel<<<B_, DO_, 0, stream>>>(s2, nullptr, out_v);                    // v = v2
}